// PPYoloELoss_26525718020732
// MI455X (gfx1250) — compile-verified
//
#include <hip/hip_runtime.h>
#include <math.h>

// ---------------- problem constants (from reference) ----------------
#define BB 16
#define NN 8400
#define CC 80
#define MM 60
#define RMAX 16
#define NBINS 17
#define TOPKK 13
#define EPSF 1e-9f
#define VFL_ALPHA 0.75f
#define CLS_W 1.0f
#define IOU_W 2.5f
#define DFL_W 0.5f

typedef __attribute__((ext_vector_type(2))) float v2f;
typedef __attribute__((ext_vector_type(8))) float v8f;

// ---------------- wave32 sum via V_WMMA_F32_16X16X4_F32 -------------
// A: lane L<16 -> (M=L,K=0)=v (vgpr0), (M=L,K=1)=0 (vgpr1)
//    lane L>=16 -> (M=L-16,K=2)=v, K=3=0
// B: all ones (4x16). D[i][j] = v_i + v_{i+16} (row sum), replicated over j.
// Lane L<16 holds D rows 0..7 in its 8 D vgprs; lane L>=16 rows 8..15.
// Sum of own 8 D vgprs + shfl_xor(16) = full 32-lane sum, exact f32 adds.
__device__ __forceinline__ float wave_sum_wmma(float v) {
    v2f a; a[0] = v;    a[1] = 0.0f;
    v2f o; o[0] = 1.0f; o[1] = 1.0f;
    v8f c = {};
    c = __builtin_amdgcn_wmma_f32_16x16x4_f32(false, a, false, o,
                                              (short)0, c, false, false);
    float s = c[0] + c[1] + c[2] + c[3] + c[4] + c[5] + c[6] + c[7];
    s += __shfl_xor(s, 16, 32);
    return s;
}

// block reduction (result valid on thread 0). EXEC all-ones: callers invoke
// this unconditionally with full blocks.
__device__ __forceinline__ float block_sum(float v, float* sbuf) {
    __syncthreads();
    float w = wave_sum_wmma(v);
    int lane = threadIdx.x & 31;
    int wid  = threadIdx.x >> 5;
    if (lane == 0) sbuf[wid] = w;
    __syncthreads();
    float r = 0.f;
    if (threadIdx.x == 0) {
        int nw = (blockDim.x + 31) >> 5;
        for (int i = 0; i < nw; ++i) r += sbuf[i];
    }
    return r;
}

__device__ __forceinline__ float iou_xyxy(float px1, float py1, float px2, float py2,
                                          float tx1, float ty1, float tx2, float ty2) {
    float ix1 = fmaxf(px1, tx1), iy1 = fmaxf(py1, ty1);
    float ix2 = fminf(px2, tx2), iy2 = fminf(py2, ty2);
    float inter = fmaxf(ix2 - ix1, 0.f) * fmaxf(iy2 - iy1, 0.f);
    float a1 = (px2 - px1) * (py2 - py1);
    float a2 = (tx2 - tx1) * (ty2 - ty1);
    float uni = a1 + a2 - inter;
    return inter / (uni + EPSF);
}

// ---------------- K1: DFL decode -> pred_boxes ----------------------
__global__ void k_decode(const float* __restrict__ pd, const float* __restrict__ ap,
                         float* __restrict__ boxes) {
    int i = blockIdx.x * blockDim.x + threadIdx.x;   // (b,n) flat
    if (i >= BB * NN) return;
    int n = i % NN;
    const float* p = pd + (size_t)i * 4 * NBINS;
    float d[4];
#pragma unroll
    for (int s = 0; s < 4; ++s) {
        const float* q = p + s * NBINS;
        float mx = q[0];
#pragma unroll
        for (int k = 1; k < NBINS; ++k) mx = fmaxf(mx, q[k]);
        float den = 0.f, num = 0.f;
#pragma unroll
        for (int k = 0; k < NBINS; ++k) {
            float e = expf(q[k] - mx);
            den += e; num += e * (float)k;
        }
        d[s] = num / den;
    }
    float ax = ap[n * 2], ay = ap[n * 2 + 1];
    float* o = boxes + (size_t)i * 4;
    o[0] = ax - d[0]; o[1] = ay - d[1]; o[2] = ax + d[2]; o[3] = ay + d[3];
}

// ---------------- K2: per-GT metric + top-13 + scatter --------------
__global__ void __launch_bounds__(256)
k_assign(const float* __restrict__ ps, const float* __restrict__ boxes,
         const float* __restrict__ ap, const int* __restrict__ gl,
         const float* __restrict__ gb, const float* __restrict__ mg,
         unsigned int* __restrict__ ascore, unsigned long long* __restrict__ bestm) {
    __shared__ float smet[NN];     // 33.6 KB, fits easily in WGP LDS
    __shared__ float rv[256];
    __shared__ int   ri[256];
    __shared__ float selv[TOPKK];
    __shared__ int   seli[TOPKK];

    int bm = blockIdx.x;
    int b = bm / MM, m = bm % MM;
    if (mg[bm] <= 0.f) return;                       // uniform per block
    int label = gl[bm]; label = min(max(label, 0), CC - 1);
    float tx1 = gb[bm * 4 + 0], ty1 = gb[bm * 4 + 1];
    float tx2 = gb[bm * 4 + 2], ty2 = gb[bm * 4 + 3];
    int tid = threadIdx.x;

    for (int n = tid; n < NN; n += 256) {
        float ax = ap[n * 2], ay = ap[n * 2 + 1];
        float mind = fminf(fminf(ax - tx1, ay - ty1), fminf(tx2 - ax, ty2 - ay));
        float met = 0.f;
        if (mind > EPSF) {
            const float* pb = boxes + ((size_t)(b * NN + n)) * 4;
            float iou = iou_xyxy(pb[0], pb[1], pb[2], pb[3], tx1, ty1, tx2, ty2);
            float x   = ps[((size_t)(b * NN + n)) * CC + label];
            float sig = 1.f / (1.f + expf(-x));
            float i2 = iou * iou;
            met = sig * (i2 * i2 * i2);              // s^1 * iou^6
        }
        smet[n] = met;
    }
    __syncthreads();

    // exact top-13 (descending, lowest index wins ties, like lax.top_k)
    for (int k = 0; k < TOPKK; ++k) {
        float bv = -1.f; int bi = 0x7fffffff;
        for (int n = tid; n < NN; n += 256) {
            float v = smet[n];
            if (v > bv) { bv = v; bi = n; }
        }
        rv[tid] = bv; ri[tid] = bi; __syncthreads();
        for (int s = 128; s > 0; s >>= 1) {
            if (tid < s) {
                float ov = rv[tid + s]; int oi = ri[tid + s];
                if (ov > rv[tid] || (ov == rv[tid] && oi < ri[tid])) {
                    rv[tid] = ov; ri[tid] = oi;
                }
            }
            __syncthreads();
        }
        if (tid == 0) { selv[k] = rv[0]; seli[k] = ri[0]; smet[ri[0]] = -2.f; }
        __syncthreads();
    }

    if (tid == 0) {
        float maxmet = selv[0];
        if (maxmet > 0.f) {
            float maxiou = 0.f;
            for (int k = 0; k < TOPKK; ++k) {
                if (selv[k] > 0.f) {
                    const float* pb = boxes + ((size_t)(b * NN + seli[k])) * 4;
                    maxiou = fmaxf(maxiou,
                        iou_xyxy(pb[0], pb[1], pb[2], pb[3], tx1, ty1, tx2, ty2));
                }
            }
            for (int k = 0; k < TOPKK; ++k) {
                if (selv[k] > 0.f) {
                    int n = seli[k];
                    float norm = selv[k] / (maxmet + EPSF) * maxiou;
                    // norms >= 0: uint compare of IEEE bits == float compare
                    atomicMax(&ascore[((size_t)(b * NN + n)) * CC + label],
                              __float_as_uint(norm));
                    // argmax_m of metric; on ties smaller m wins (0xFFFF - m)
                    unsigned long long packed =
                        ((unsigned long long)__float_as_uint(selv[k]) << 32) |
                        (unsigned long long)(0xFFFFu - (unsigned)m);
                    atomicMax(&bestm[b * NN + n], packed);
                }
            }
        }
    }
}

// ---------------- K2b: assigned label per anchor --------------------
__global__ void k_labels(const unsigned long long* __restrict__ bestm,
                         const int* __restrict__ gl, int* __restrict__ albl) {
    int i = blockIdx.x * blockDim.x + threadIdx.x;
    if (i >= BB * NN) return;
    unsigned long long p = bestm[i];
    int lab = -1;
    if (p != 0ull) {
        int b = i / NN;
        unsigned m = 0xFFFFu - (unsigned)(p & 0xFFFFFFFFull);
        int l = gl[b * MM + (int)m];
        lab = min(max(l, 0), CC - 1);
    }
    albl[i] = lab;
}

// ---------------- K3: varifocal cls loss + sum(a_scores) ------------
__global__ void __launch_bounds__(256)
k_cls(const float* __restrict__ ps, const unsigned int* __restrict__ ascore,
      const int* __restrict__ albl, double* __restrict__ acc) {
    __shared__ float sbuf[8];
    const size_t T = (size_t)BB * NN * CC;
    float lcls = 0.f, lq = 0.f;
    for (size_t i = blockIdx.x * (size_t)blockDim.x + threadIdx.x; i < T;
         i += (size_t)gridDim.x * blockDim.x) {
        float x = ps[i];
        float q = __uint_as_float(ascore[i]);
        int bn = (int)(i / CC);
        int c  = (int)(i % CC);
        float sig = 1.f / (1.f + expf(-x));
        float w = (c == albl[bn]) ? q : (VFL_ALPHA * sig * sig);
        float bce = fmaxf(x, 0.f) - x * q + log1pf(expf(-fabsf(x)));
        lcls += w * bce;
        lq   += q;
    }
    float r0 = block_sum(lcls, sbuf);
    float r1 = block_sum(lq, sbuf);
    if (threadIdx.x == 0) {
        atomicAdd(&acc[0], (double)r0);
        atomicAdd(&acc[1], (double)r1);
    }
}

// ---------------- K4: GIoU + DFL over positives ---------------------
__global__ void __launch_bounds__(256)
k_pos(const float* __restrict__ pd, const float* __restrict__ boxes,
      const float* __restrict__ ap, const float* __restrict__ gb,
      const unsigned long long* __restrict__ bestm, double* __restrict__ acc) {
    __shared__ float sbuf[8];
    int i = blockIdx.x * blockDim.x + threadIdx.x;
    float liou = 0.f, ldfl = 0.f, lnp = 0.f;
    if (i < BB * NN) {
        unsigned long long p = bestm[i];
        if (p != 0ull) {
            int b = i / NN, n = i % NN;
            unsigned m = 0xFFFFu - (unsigned)(p & 0xFFFFFFFFull);
            const float* t  = gb + ((size_t)(b * MM) + m) * 4;
            const float* pb = boxes + (size_t)i * 4;
            float tx1 = t[0], ty1 = t[1], tx2 = t[2], ty2 = t[3];
            float px1 = pb[0], py1 = pb[1], px2 = pb[2], py2 = pb[3];
            float ix1 = fmaxf(px1, tx1), iy1 = fmaxf(py1, ty1);
            float ix2 = fminf(px2, tx2), iy2 = fminf(py2, ty2);
            float inter = fmaxf(ix2 - ix1, 0.f) * fmaxf(iy2 - iy1, 0.f);
            float a1 = (px2 - px1) * (py2 - py1);
            float a2 = (tx2 - tx1) * (ty2 - ty1);
            float uni = a1 + a2 - inter;
            float iou = inter / (uni + 1e-9f);
            float ex1 = fminf(px1, tx1), ey1 = fminf(py1, ty1);
            float ex2 = fmaxf(px2, tx2), ey2 = fmaxf(py2, ty2);
            float enc = (ex2 - ex1) * (ey2 - ey1);
            float giou = iou - (enc - uni) / (enc + 1e-9f);
            liou = 1.f - giou;
            lnp  = 1.f;
            // DFL
            float ax = ap[n * 2], ay = ap[n * 2 + 1];
            float tv[4] = {ax - tx1, ay - ty1, tx2 - ax, ty2 - ay};
            const float* q = pd + (size_t)i * 4 * NBINS;
#pragma unroll
            for (int s = 0; s < 4; ++s) {
                float tc = fminf(fmaxf(tv[s], 0.f), (float)RMAX - 0.01f);
                int tl = (int)floorf(tc); tl = min(max(tl, 0), RMAX);
                int tr = min(tl + 1, RMAX);
                float wr = tc - (float)tl, wl = 1.f - wr;
                const float* qq = q + s * NBINS;
                float mx = qq[0];
#pragma unroll
                for (int k = 1; k < NBINS; ++k) mx = fmaxf(mx, qq[k]);
                float den = 0.f;
#pragma unroll
                for (int k = 0; k < NBINS; ++k) den += expf(qq[k] - mx);
                float lden = logf(den);
                ldfl += -(qq[tl] - mx - lden) * wl - (qq[tr] - mx - lden) * wr;
            }
        }
    }
    float r0 = block_sum(liou, sbuf);
    float r1 = block_sum(ldfl, sbuf);
    float r2 = block_sum(lnp,  sbuf);
    if (threadIdx.x == 0) {
        atomicAdd(&acc[2], (double)r0);
        atomicAdd(&acc[3], (double)r1);
        atomicAdd(&acc[4], (double)r2);
    }
}

// ---------------- K5: combine scalars -------------------------------
__global__ void k_final(const double* __restrict__ acc, float* __restrict__ out) {
    if (threadIdx.x == 0 && blockIdx.x == 0) {
        double qs = acc[1]; if (qs < 1.0) qs = 1.0;
        double np = acc[4]; if (np < 1.0) np = 1.0;
        double loss = (double)CLS_W * acc[0] / qs
                    + (double)IOU_W * acc[2] / np
                    + (double)DFL_W * acc[3] / (np * 4.0);
        out[0] = (float)loss;
    }
}

extern "C" void kernel_launch(void* const* d_in, const int* in_sizes, int n_in,
                              void* d_out, int out_size, void* d_ws, size_t ws_size,
                              hipStream_t stream) {
    const float* ps = (const float*)d_in[0];   // pred_scores  [B,N,C]
    const float* pd = (const float*)d_in[1];   // pred_dist    [B,N,68]
    const float* ap = (const float*)d_in[2];   // anchor_points[N,2]
    const int*   gl = (const int*)  d_in[3];   // gt_labels    [B,M,1]
    const float* gb = (const float*)d_in[4];   // gt_bboxes    [B,M,4]
    const float* mg = (const float*)d_in[5];   // mask_gt      [B,M,1]
    float* out = (float*)d_out;

    char* ws = (char*)d_ws;
    size_t off = 0;
    float* boxes = (float*)(ws + off);
    off += (size_t)BB * NN * 4 * sizeof(float);                 // 2,150,400 B
    size_t zero_from = off;
    unsigned int* ascore = (unsigned int*)(ws + off);
    off += (size_t)BB * NN * CC * sizeof(unsigned int);         // 43,008,000 B
    unsigned long long* bestm = (unsigned long long*)(ws + off);
    off += (size_t)BB * NN * sizeof(unsigned long long);        // 1,075,200 B
    int* albl = (int*)(ws + off);
    off += (size_t)BB * NN * sizeof(int);                       // 537,600 B
    double* acc = (double*)(ws + off);
    off += 8 * sizeof(double);

    hipMemsetAsync(ws + zero_from, 0, off - zero_from, stream);

    int bnBlocks = (BB * NN + 255) / 256;   // 525 blocks, exact fit
    k_decode<<<bnBlocks, 256, 0, stream>>>(pd, ap, boxes);
    k_assign<<<BB * MM, 256, 0, stream>>>(ps, boxes, ap, gl, gb, mg, ascore, bestm);
    k_labels<<<bnBlocks, 256, 0, stream>>>(bestm, gl, albl);
    k_cls<<<2048, 256, 0, stream>>>(ps, ascore, albl, acc);
    k_pos<<<bnBlocks, 256, 0, stream>>>(pd, boxes, ap, gb, bestm, acc);
    k_final<<<1, 1, 0, stream>>>(acc, out);
}